// WordSeqAttentionModel_23467701306003
// MI455X (gfx1250) — compile-verified
//
#include <hip/hip_runtime.h>

typedef float v2f __attribute__((ext_vector_type(2)));
typedef float v8f __attribute__((ext_vector_type(8)));

constexpr int kB = 64;
constexpr int kL = 2048;
constexpr int kD = 512;

// ---------------------------------------------------------------------------
// Kernel 0: zero the accumulator region of the workspace (num + den).
// ---------------------------------------------------------------------------
__global__ void zero_kernel(float* __restrict__ p, int n) {
    int i = blockIdx.x * blockDim.x + threadIdx.x;
    if (i < n) p[i] = 0.0f;
}

// ---------------------------------------------------------------------------
// Kernel 1: x_w = x @ W.T + b_lin  via V_WMMA_F32_16X16X4_F32.
// One wave per 16x16 output tile; K loop over 512 in steps of 4.
// A (16x4 f32) lane layout: lanes 0-15 -> M=0..15 / K={0,1}; lanes 16-31 -> K={2,3}.
// B (4x16 f32) mirrored: lane n holds column j0+n, K pair selected by lane half.
// C/D (16x16 f32): VGPR r holds M=r (lanes 0-15) and M=r+8 (lanes 16-31), N=lane%16.
// ---------------------------------------------------------------------------
__global__ __launch_bounds__(32) void xw_wmma_kernel(const float* __restrict__ x,
                                                     const float* __restrict__ W,
                                                     const float* __restrict__ b_lin,
                                                     float* __restrict__ xw) {
    const int lane = threadIdx.x;      // 0..31
    const int half = lane >> 4;        // 0 or 1 (selects K pair)
    const int m    = lane & 15;        // row within tile (A) / col within tile (B)
    const int i0   = (blockIdx.x & 3) << 4;   // 4 tiles over B=64 rows
    const int j0   = (blockIdx.x >> 2) << 4;  // 32 tiles over D=512 cols

    const float* __restrict__ arow = x + (size_t)(i0 + m) * kD;  // x[i0+m, :]
    const float* __restrict__ brow = W + (size_t)(j0 + m) * kD;  // W[j0+m, :] (== W.T column)

    v8f c = {0.f, 0.f, 0.f, 0.f, 0.f, 0.f, 0.f, 0.f};
    #pragma unroll 8
    for (int k = 0; k < kD; k += 4) {
        v2f a, b;
        a.x = arow[k + 2 * half];
        a.y = arow[k + 2 * half + 1];
        b.x = brow[k + 2 * half];
        b.y = brow[k + 2 * half + 1];
        // 8 args: (neg_a, A, neg_b, B, c_mod, C, reuse_a, reuse_b)
        c = __builtin_amdgcn_wmma_f32_16x16x4_f32(false, a, false, b,
                                                  (short)0, c, false, false);
    }

    const int mbase = half << 3;          // lanes 16-31 hold rows M=8..15
    const float bl  = b_lin[j0 + m];
    #pragma unroll
    for (int r = 0; r < 8; ++r)
        xw[(size_t)(i0 + mbase + r) * kD + j0 + m] = c[r] + bl;
}

// ---------------------------------------------------------------------------
// Kernel 2: single fused pass over seq (read once).
// Grid: (16 chunks of L) x (64 batches); 256 threads = 8 waves; 16 rows/wave.
// Per row: coalesced float4 loads, dot with x_w (kept in regs), wave32 xor
// reduction, score = l<len ? tanh(dot+bias) : -1, e = exp(score).
// Accumulate e (den) and e*row (num) without re-reading seq.
// ---------------------------------------------------------------------------
__global__ __launch_bounds__(256) void fused_kernel(const float* __restrict__ seq,
                                                    const float* __restrict__ xw,
                                                    const int*   __restrict__ lengths,
                                                    const float* __restrict__ bias,
                                                    float* __restrict__ wout,   // unnormalized e
                                                    float* __restrict__ num,
                                                    float* __restrict__ den) {
    __shared__ float num_s[kD];
    __shared__ float den_s;

    const int tid = threadIdx.x;
    const int b   = blockIdx.y;

    num_s[tid]       = 0.0f;
    num_s[tid + 256] = 0.0f;
    if (tid == 0) den_s = 0.0f;
    __syncthreads();

    const int wave  = tid >> 5;
    const int lane  = tid & 31;
    const int lbase = blockIdx.x * 128 + wave * 16;
    const int len   = lengths[b];
    const float bias0 = bias[0];

    // x_w row for this batch: lane owns d = lane*4 + 128*j + c  (j=0..3, c=0..3)
    const float4* __restrict__ xw4 = (const float4*)(xw + (size_t)b * kD);
    float4 xwv[4];
    #pragma unroll
    for (int j = 0; j < 4; ++j) xwv[j] = xw4[lane + 32 * j];

    float4 nl[4];
    #pragma unroll
    for (int j = 0; j < 4; ++j) nl[j] = make_float4(0.f, 0.f, 0.f, 0.f);
    float denloc = 0.0f;

    for (int r = 0; r < 16; ++r) {
        const int l = lbase + r;
        const float4* __restrict__ row4 =
            (const float4*)(seq + ((size_t)b * kL + l) * kD);

        float4 rv[4];
        float acc = 0.0f;
        #pragma unroll
        for (int j = 0; j < 4; ++j) {
            rv[j] = row4[lane + 32 * j];            // global_load_b128, coalesced
            acc += rv[j].x * xwv[j].x + rv[j].y * xwv[j].y +
                   rv[j].z * xwv[j].z + rv[j].w * xwv[j].w;
        }
        // wave32 butterfly reduction -> all lanes hold full dot product
        #pragma unroll
        for (int off = 16; off > 0; off >>= 1)
            acc += __shfl_xor(acc, off, 32);

        // mask BEFORE tanh: tanh(-inf) = -1 (masked rows still get weight e^-1)
        const float score = (l < len) ? tanhf(acc + bias0) : -1.0f;
        const float e = __expf(score);   // scores in [-1,1]: no max-subtract needed

        if (lane == 0) wout[(size_t)b * kL + l] = e;
        denloc += e;                     // identical on all lanes; lane 0 flushes
        #pragma unroll
        for (int j = 0; j < 4; ++j) {
            nl[j].x += e * rv[j].x;
            nl[j].y += e * rv[j].y;
            nl[j].z += e * rv[j].z;
            nl[j].w += e * rv[j].w;
        }
    }

    // cross-wave reduction via LDS float atomics (ds_add_f32)
    #pragma unroll
    for (int j = 0; j < 4; ++j) {
        const int d0 = lane * 4 + 128 * j;
        atomicAdd(&num_s[d0 + 0], nl[j].x);
        atomicAdd(&num_s[d0 + 1], nl[j].y);
        atomicAdd(&num_s[d0 + 2], nl[j].z);
        atomicAdd(&num_s[d0 + 3], nl[j].w);
    }
    if (lane == 0) atomicAdd(&den_s, denloc);
    __syncthreads();

    // cross-workgroup reduction via global f32 atomics
    atomicAdd(&num[(size_t)b * kD + tid],       num_s[tid]);
    atomicAdd(&num[(size_t)b * kD + 256 + tid], num_s[tid + 256]);
    if (tid == 0) atomicAdd(&den[b], den_s);
}

// ---------------------------------------------------------------------------
// Kernel 3: normalize. hidden = num/den; weight = e/den (in place on d_out).
// ---------------------------------------------------------------------------
__global__ void finalize_kernel(const float* __restrict__ num,
                                const float* __restrict__ den,
                                float* __restrict__ hidden,
                                float* __restrict__ wout) {
    const int i = blockIdx.x * 256 + threadIdx.x;
    if (i < kB * kD) {
        const int b = i >> 9;            // / 512
        hidden[i] = num[i] / den[b];
    } else if (i < kB * kD + kB * kL) {
        const int k = i - kB * kD;
        const int b = k >> 11;           // / 2048
        wout[k] = wout[k] / den[b];
    }
}

// ---------------------------------------------------------------------------
extern "C" void kernel_launch(void* const* d_in, const int* in_sizes, int n_in,
                              void* d_out, int out_size, void* d_ws, size_t ws_size,
                              hipStream_t stream) {
    const float* x       = (const float*)d_in[0];   // (64, 512)
    const float* seq     = (const float*)d_in[1];   // (64, 2048, 512)
    const int*   lengths = (const int*)  d_in[2];   // (64,)
    const float* W       = (const float*)d_in[3];   // (512, 512)
    const float* b_lin   = (const float*)d_in[4];   // (512,)
    const float* bias    = (const float*)d_in[5];   // (1,)

    float* hidden = (float*)d_out;            // (64, 512)
    float* wout   = hidden + kB * kD;         // (64, 2048)

    float* xw  = (float*)d_ws;                // (64, 512)
    float* num = xw + kB * kD;                // (64, 512)
    float* den = num + kB * kD;               // (64,)

    // accumulators must be zeroed every call (graph replay reuses buffers)
    zero_kernel<<<(kB * kD + kB + 255) / 256, 256, 0, stream>>>(num, kB * kD + kB);

    xw_wmma_kernel<<<128, 32, 0, stream>>>(x, W, b_lin, xw);

    fused_kernel<<<dim3(16, kB), 256, 0, stream>>>(seq, xw, lengths, bias,
                                                   wout, num, den);

    finalize_kernel<<<(kB * kD + kB * kL + 255) / 256, 256, 0, stream>>>(
        num, den, hidden, wout);
}